// MANO_47605417509443
// MI455X (gfx1250) — compile-verified
//
#include <hip/hip_runtime.h>

// ---------------- problem constants ----------------
#define Bn    16384
#define Vn    778
#define Jn    16
#define NBn   10
#define NPCAn 6
#define N3    (Vn * 3)      // 2334 live columns of the big GEMM
#define KPAD  148           // [betas(10) | pad(2) | pose_feature(135) | pad(1)]
#define KB4   (KPAD / 4)    // 37 k-blocks of 4
#define NP    2368          // padded columns: 37 groups * 64
#define VP    780           // V padded to multiple of 4

typedef float v2f __attribute__((ext_vector_type(2)));
typedef float v4f __attribute__((ext_vector_type(4)));
typedef float v8f __attribute__((ext_vector_type(8)));

__device__ __forceinline__ v8f wmma_f32_16x16x4(v2f a, v2f b, v8f c) {
  // D = A(16x4 f32) * B(4x16 f32) + C(16x16 f32)
  return __builtin_amdgcn_wmma_f32_16x16x4_f32(
      /*neg_a=*/false, a, /*neg_b=*/false, b,
      /*c_mod=*/(short)0, c, /*reuse_a=*/false, /*reuse_b=*/false);
}

// ---------------------------------------------------------------------------
// Kernel 0: pack the GEMM B matrix (shapedirs rows 0..9, posedirs rows 12..146,
// zero pads) into Bpack[kb4][n][4] so the WMMA loop does unconditional b64
// loads. Also zero-pad J_regressor to 16 x 780.
// ---------------------------------------------------------------------------
__global__ void mano_pack_kernel(const float* __restrict__ sd,
                                 const float* __restrict__ pd,
                                 const float* __restrict__ jreg,
                                 float* __restrict__ bpack,
                                 float* __restrict__ jregp) {
  int idx = blockIdx.x * blockDim.x + threadIdx.x;
  const int NB_ENTRIES = KB4 * NP;       // 87,616 packed 4-float entries
  if (idx < NB_ENTRIES) {
    int kb4 = idx / NP;
    int n = idx % NP;
    v4f vals;
#pragma unroll
    for (int kk = 0; kk < 4; ++kk) {
      int k = kb4 * 4 + kk;
      float v = 0.0f;
      if (n < N3) {
        if (k < NBn) v = sd[n * NBn + k];                       // betas part
        else if (k >= 12 && k < 147) v = pd[(size_t)(k - 12) * N3 + n];  // pose part
      }
      vals[kk] = v;
    }
    *(v4f*)(bpack + (size_t)idx * 4) = vals;
  } else {
    int j = idx - NB_ENTRIES;
    if (j < Jn * VP) {
      int jj = j / VP;
      int v = j % VP;
      jregp[j] = (v < Vn) ? jreg[jj * Vn + v] : 0.0f;
    }
  }
}

// ---------------------------------------------------------------------------
// Kernel 1: per-batch pose math.
//   full_pose = [global_orient, hand_pose @ hand_components] + pose_mean
//   rot_mats  = rodrigues(full_pose)            -> rotws  (B x 16 x 9)
//   afeat     = [betas, 0,0, rot[1:]-I, 0]      -> afeat  (B x 148), WMMA A rows
// ---------------------------------------------------------------------------
__global__ void mano_pose_kernel(const float* __restrict__ go,
                                 const float* __restrict__ hp,
                                 const float* __restrict__ hc,
                                 const float* __restrict__ pm,
                                 const float* __restrict__ betas,
                                 float* __restrict__ afeat,
                                 float* __restrict__ rotws) {
  int b = blockIdx.x * blockDim.x + threadIdx.x;
  if (b >= Bn) return;

  float fp[48];
  fp[0] = go[b * 3 + 0] + pm[0];
  fp[1] = go[b * 3 + 1] + pm[1];
  fp[2] = go[b * 3 + 2] + pm[2];
  float hpv[NPCAn];
#pragma unroll
  for (int p = 0; p < NPCAn; ++p) hpv[p] = hp[b * NPCAn + p];
#pragma unroll
  for (int c = 0; c < 45; ++c) {
    float acc = pm[3 + c];
#pragma unroll
    for (int p = 0; p < NPCAn; ++p) acc += hpv[p] * hc[p * 45 + c];
    fp[3 + c] = acc;
  }

  float* af = afeat + (size_t)b * KPAD;
#pragma unroll
  for (int l = 0; l < NBn; ++l) af[l] = betas[b * NBn + l];
  af[10] = 0.0f;
  af[11] = 0.0f;
  af[147] = 0.0f;

  float* rw = rotws + (size_t)b * 144;
#pragma unroll
  for (int j = 0; j < Jn; ++j) {
    float rx = fp[3 * j + 0], ry = fp[3 * j + 1], rz = fp[3 * j + 2];
    float ax = rx + 1e-8f, ay = ry + 1e-8f, az = rz + 1e-8f;
    float ang = sqrtf(ax * ax + ay * ay + az * az);
    float inv = 1.0f / ang;
    float ux = rx * inv, uy = ry * inv, uz = rz * inv;
    float s = sinf(ang);
    float c1 = 1.0f - cosf(ang);
    float R[9];
    R[0] = 1.0f - c1 * (uy * uy + uz * uz);
    R[1] = -s * uz + c1 * ux * uy;
    R[2] = s * uy + c1 * ux * uz;
    R[3] = s * uz + c1 * ux * uy;
    R[4] = 1.0f - c1 * (ux * ux + uz * uz);
    R[5] = -s * ux + c1 * uy * uz;
    R[6] = -s * uy + c1 * ux * uz;
    R[7] = s * ux + c1 * uy * uz;
    R[8] = 1.0f - c1 * (ux * ux + uy * uy);
#pragma unroll
    for (int e = 0; e < 9; ++e) rw[j * 9 + e] = R[e];
    if (j > 0) {
#pragma unroll
      for (int e = 0; e < 9; ++e)
        af[12 + (j - 1) * 9 + e] = R[e] - ((e == 0 || e == 4 || e == 8) ? 1.0f : 0.0f);
    }
  }
}

// ---------------------------------------------------------------------------
// Kernel 2: v_posed = v_template + [betas|pose_feature] @ Bpack
// Pure WMMA main loop: 1 A b64 load + 4 B b64 loads + 4 v_wmma per k-step,
// no divergent guards. Each wave: 1 batch-tile x 4 column tiles.
// Result written straight into d_out's vertex region.
// ---------------------------------------------------------------------------
__global__ void __launch_bounds__(256) mano_vposed_kernel(
    const float* __restrict__ afeat, const float* __restrict__ bpack,
    const float* __restrict__ vt, float* __restrict__ vout) {
  const int NGRP = 37;
  int wid = (blockIdx.x * blockDim.x + threadIdx.x) >> 5;
  int lane = threadIdx.x & 31;
  int btile = wid / NGRP;
  int ngrp = wid % NGRP;
  if (btile >= Bn / 16) return;

  int m = lane & 15;       // A row within tile / B-D column within tile
  int khalf = lane >> 4;   // K/M half select
  int n0 = ngrp * 64;

  const float* arow = afeat + (size_t)(btile * 16 + m) * KPAD + 2 * khalf;
  v8f acc[4] = {v8f{}, v8f{}, v8f{}, v8f{}};

  for (int s = 0; s < KB4; ++s) {
    v2f a = *(const v2f*)(arow + s * 4);
#pragma unroll
    for (int t = 0; t < 4; ++t) {
      int n = n0 + t * 16 + m;
      v2f bf = *(const v2f*)(bpack + ((size_t)s * NP + n) * 4 + 2 * khalf);
      acc[t] = wmma_f32_16x16x4(a, bf, acc[t]);
    }
  }

  // Epilogue: + v_template (broadcast over batch rows); lane column is fixed.
#pragma unroll
  for (int t = 0; t < 4; ++t) {
    int n = n0 + t * 16 + m;
    if (n < N3) {
      float tv = vt[n];
#pragma unroll
      for (int r = 0; r < 8; ++r) {
        int row = btile * 16 + 8 * khalf + r;
        vout[(size_t)row * N3 + n] = acc[t][r] + tv;
      }
    }
  }
}

// ---------------------------------------------------------------------------
// Kernel 3: joints[b,j,k] = sum_v J_regressor[j,v] * v_shaped[b,v,k]
// WMMA: M = 16 batch rows, N = 16 joints, K = V padded to 780.
// B loads unconditional (zero-padded jregp); A recomputed on the fly with a
// clamped (branchless) index -- padded-zero B rows annihilate the tail.
// ---------------------------------------------------------------------------
__global__ void __launch_bounds__(256) mano_joints_kernel(
    const float* __restrict__ betas, const float* __restrict__ sd,
    const float* __restrict__ vt, const float* __restrict__ jregp,
    float* __restrict__ jnts) {
  int wid = (blockIdx.x * blockDim.x + threadIdx.x) >> 5;
  int lane = threadIdx.x & 31;
  if (wid >= (Bn / 16) * 3) return;
  int btile = wid / 3;
  int kc = wid % 3;

  int m = lane & 15;
  int khalf = lane >> 4;
  int brow = btile * 16 + m;

  float bet[NBn];
#pragma unroll
  for (int l = 0; l < NBn; ++l) bet[l] = betas[brow * NBn + l];

  v8f acc = v8f{};
  for (int vb = 0; vb < VP; vb += 4) {
    v2f a, bf;
#pragma unroll
    for (int r = 0; r < 2; ++r) {
      int v = vb + 2 * khalf + r;
      int vc = (v < Vn) ? v : (Vn - 1);   // branchless clamp
      int col = vc * 3 + kc;
      float av = vt[col];
#pragma unroll
      for (int l = 0; l < NBn; ++l) av += bet[l] * sd[col * NBn + l];
      float bv = jregp[m * VP + v];       // lane's B column N = m = joint index
      if (r == 0) { a.x = av; bf.x = bv; } else { a.y = av; bf.y = bv; }
    }
    acc = wmma_f32_16x16x4(a, bf, acc);
  }

#pragma unroll
  for (int r = 0; r < 8; ++r) {
    int row = btile * 16 + 8 * khalf + r;          // batch row (C M-index)
    jnts[(size_t)row * 48 + m * 3 + kc] = acc[r];  // joint = C N-index = m
  }
}

// ---------------------------------------------------------------------------
// Kernel 4: kinematic chain + rel_transforms (3x4), posed_joints -> d_out.
// parents hard-coded (matches reference PARENTS) so everything unrolls into
// registers.
// ---------------------------------------------------------------------------
__global__ void mano_chain_kernel(const float* __restrict__ rotws,
                                  const float* __restrict__ jntws,
                                  const float* __restrict__ transl,
                                  float* __restrict__ relws,
                                  float* __restrict__ jout) {
  constexpr int PAR[16] = {-1, 0, 1, 2, 0, 4, 5, 0, 7, 8, 0, 10, 11, 0, 13, 14};
  int b = blockIdx.x * blockDim.x + threadIdx.x;
  if (b >= Bn) return;

  float jl[16][3];
#pragma unroll
  for (int j = 0; j < Jn; ++j) {
#pragma unroll
    for (int k = 0; k < 3; ++k) jl[j][k] = jntws[(size_t)b * 48 + j * 3 + k];
  }
  const float* rw = rotws + (size_t)b * 144;

  float ch[16][12];
#pragma unroll
  for (int row = 0; row < 3; ++row) {
#pragma unroll
    for (int col = 0; col < 3; ++col) ch[0][row * 4 + col] = rw[row * 3 + col];
    ch[0][row * 4 + 3] = jl[0][row];
  }
#pragma unroll
  for (int i = 1; i < Jn; ++i) {
    const int p = PAR[i];
    float rj0 = jl[i][0] - jl[p][0];
    float rj1 = jl[i][1] - jl[p][1];
    float rj2 = jl[i][2] - jl[p][2];
    const float* Ri = rw + i * 9;
#pragma unroll
    for (int row = 0; row < 3; ++row) {
#pragma unroll
      for (int col = 0; col < 3; ++col) {
        ch[i][row * 4 + col] = ch[p][row * 4 + 0] * Ri[0 * 3 + col] +
                               ch[p][row * 4 + 1] * Ri[1 * 3 + col] +
                               ch[p][row * 4 + 2] * Ri[2 * 3 + col];
      }
      ch[i][row * 4 + 3] = ch[p][row * 4 + 0] * rj0 + ch[p][row * 4 + 1] * rj1 +
                           ch[p][row * 4 + 2] * rj2 + ch[p][row * 4 + 3];
    }
  }

  float tx = transl[b * 3 + 0], ty = transl[b * 3 + 1], tz = transl[b * 3 + 2];
  float* rel = relws + (size_t)b * 192;
#pragma unroll
  for (int i = 0; i < Jn; ++i) {
    jout[((size_t)b * Jn + i) * 3 + 0] = ch[i][3] + tx;
    jout[((size_t)b * Jn + i) * 3 + 1] = ch[i][7] + ty;
    jout[((size_t)b * Jn + i) * 3 + 2] = ch[i][11] + tz;
    float jx = jl[i][0], jy = jl[i][1], jz = jl[i][2];
#pragma unroll
    for (int row = 0; row < 3; ++row) {
      float ib = ch[i][row * 4 + 0] * jx + ch[i][row * 4 + 1] * jy +
                 ch[i][row * 4 + 2] * jz;
      rel[i * 12 + row * 4 + 0] = ch[i][row * 4 + 0];
      rel[i * 12 + row * 4 + 1] = ch[i][row * 4 + 1];
      rel[i * 12 + row * 4 + 2] = ch[i][row * 4 + 2];
      rel[i * 12 + row * 4 + 3] = ch[i][row * 4 + 3] - ib;
    }
  }
}

// ---------------------------------------------------------------------------
// Kernel 5: skinning, in place on d_out's vertex region (v_posed -> vertices).
// One block per batch element; rel_transforms staged in LDS.
// ---------------------------------------------------------------------------
__global__ void __launch_bounds__(256) mano_skin_kernel(
    const float* __restrict__ lbsw, const float* __restrict__ relws,
    const float* __restrict__ transl, float* __restrict__ vbuf) {
  int b = blockIdx.x;
  __shared__ float srel[192];
  __shared__ float str[3];
  if (threadIdx.x < 192) srel[threadIdx.x] = relws[(size_t)b * 192 + threadIdx.x];
  if (threadIdx.x < 3) str[threadIdx.x] = transl[b * 3 + threadIdx.x];
  __syncthreads();

  for (int v = threadIdx.x; v < Vn; v += 256) {
    float* vp = vbuf + ((size_t)b * Vn + v) * 3;
    float px = vp[0], py = vp[1], pz = vp[2];
    float T[12];
#pragma unroll
    for (int e = 0; e < 12; ++e) T[e] = 0.0f;
#pragma unroll
    for (int j = 0; j < Jn; ++j) {
      float w = lbsw[v * Jn + j];
#pragma unroll
      for (int e = 0; e < 12; ++e) T[e] += w * srel[j * 12 + e];
    }
    vp[0] = T[0] * px + T[1] * py + T[2] * pz + T[3] + str[0];
    vp[1] = T[4] * px + T[5] * py + T[6] * pz + T[7] + str[1];
    vp[2] = T[8] * px + T[9] * py + T[10] * pz + T[11] + str[2];
  }
}

// ---------------------------------------------------------------------------
extern "C" void kernel_launch(void* const* d_in, const int* in_sizes, int n_in,
                              void* d_out, int out_size, void* d_ws,
                              size_t ws_size, hipStream_t stream) {
  (void)in_sizes; (void)n_in; (void)out_size; (void)ws_size;

  const float* betas  = (const float*)d_in[0];
  const float* go     = (const float*)d_in[1];
  const float* hp     = (const float*)d_in[2];
  const float* transl = (const float*)d_in[3];
  const float* hc     = (const float*)d_in[4];
  const float* pm     = (const float*)d_in[5];
  const float* vt     = (const float*)d_in[6];
  const float* sd     = (const float*)d_in[7];
  const float* pd     = (const float*)d_in[8];
  const float* jreg   = (const float*)d_in[9];
  const float* lbsw   = (const float*)d_in[10];
  // d_in[11] (parents) is a compile-time constant in the reference; hard-coded.

  float* out   = (float*)d_out;
  float* verts = out;                              // B*V*3 (also holds v_posed)
  float* jout  = out + (size_t)Bn * Vn * 3;        // B*J*3

  float* ws    = (float*)d_ws;
  float* afeat = ws;                               // B*148
  float* rotws = afeat + (size_t)Bn * KPAD;        // B*144
  float* jntws = rotws + (size_t)Bn * 144;         // B*48
  float* relws = jntws + (size_t)Bn * 48;          // B*192
  float* bpack = relws + (size_t)Bn * 192;         // 37*2368*4
  float* jregp = bpack + (size_t)KB4 * NP * 4;     // 16*780

  // Pack B operands (GEMM B matrix + padded J_regressor).
  {
    int total = KB4 * NP + Jn * VP;
    mano_pack_kernel<<<(total + 255) / 256, 256, 0, stream>>>(sd, pd, jreg,
                                                              bpack, jregp);
  }

  mano_pose_kernel<<<Bn / 256, 256, 0, stream>>>(go, hp, hc, pm, betas, afeat,
                                                 rotws);

  // (B/16) btiles * 37 column groups = 37888 waves; 8 waves per block.
  mano_vposed_kernel<<<(Bn / 16) * 37 / 8, 256, 0, stream>>>(afeat, bpack, vt,
                                                             verts);

  // (B/16) btiles * 3 components = 3072 waves; 8 waves per block.
  mano_joints_kernel<<<(Bn / 16) * 3 / 8, 256, 0, stream>>>(betas, sd, vt,
                                                            jregp, jntws);

  mano_chain_kernel<<<Bn / 256, 256, 0, stream>>>(rotws, jntws, transl, relws,
                                                  jout);

  mano_skin_kernel<<<Bn, 256, 0, stream>>>(lbsw, relws, transl, verts);
}